// GTCLayer_64458869178862
// MI455X (gfx1250) — compile-verified
//
#include <hip/hip_runtime.h>
#include <hip/hip_bf16.h>
#include <math.h>

// Problem constants (match reference)
#define NN 131072
#define EE 2097152
#define DD 128
#define EFD 64
#define NGRP 64
#define GSZ 2048  // NN / NGRP

typedef __attribute__((ext_vector_type(2))) float v2f;
typedef __attribute__((ext_vector_type(8))) float v8f;
typedef __attribute__((ext_vector_type(4))) unsigned int u32x4;
typedef __attribute__((ext_vector_type(4))) int i32x4;
typedef __attribute__((ext_vector_type(8))) int i32x8;

#if defined(__has_builtin)
#if __has_builtin(__builtin_amdgcn_tensor_load_to_lds) && \
    __has_builtin(__builtin_amdgcn_s_wait_tensorcnt)
#define USE_TDM 1
#endif
#endif

// D = A(16x4 f32) * B(4x16 f32) + C(16x16 f32), full-wave op (wave32).
__device__ __forceinline__ v8f wmma4(v2f a, v2f b, v8f c) {
  // (neg_a, A, neg_b, B, c_mod, C, reuse_a, reuse_b)
  return __builtin_amdgcn_wmma_f32_16x16x4_f32(false, a, false, b, (short)0, c,
                                               false, false);
}

// ---------------------------------------------------------------------------
// Kernel 1: h_self = relu([nfeat | cos(ts*freq+phase)] @ W_time + b_time)
// One wave -> 16 rows x 128 cols. K = 256 (128 nfeat + 128 time-encoding).
// A-frag layout (ISA 16x4 f32): lanes 0-15 M=lane K={0,1}; lanes 16-31 K={2,3}
// C/D layout: vgpr r -> (M = r + 8*(lane>=16), N = lane&15).
// ---------------------------------------------------------------------------
__global__ __launch_bounds__(256) void hself_kernel(
    const float* __restrict__ nfeat, const float* __restrict__ timestamp,
    const float* __restrict__ basis_freq, const float* __restrict__ phase,
    const float* __restrict__ W_time, const float* __restrict__ b_time,
    float* __restrict__ h_self) {
  const int wave = (blockIdx.x * blockDim.x + threadIdx.x) >> 5;
  const int lane = threadIdx.x & 31;
  const int rowBase = wave * 16;
  const int mrow = rowBase + (lane & 15);
  const int kk = (lane >> 4) * 2;  // per-lane K sub-offset {0,2}
  const int ncol = lane & 15;

  v8f acc[8] = {};
  const float ts = timestamp[mrow];

  // K = 0..127 : nfeat part
  for (int k0 = 0; k0 < DD; k0 += 4) {
    v2f a;
    const float* ap = nfeat + (size_t)mrow * DD + k0 + kk;
    a.x = ap[0];
    a.y = ap[1];
#pragma unroll
    for (int t = 0; t < 8; ++t) {
      v2f b;
      const float* bp = W_time + (size_t)(k0 + kk) * DD + t * 16 + ncol;
      b.x = bp[0];
      b.y = bp[DD];
      acc[t] = wmma4(a, b, acc[t]);
    }
  }
  // K = 128..255 : time-encoding part (A computed on the fly)
  for (int k0 = 0; k0 < DD; k0 += 4) {
    const int k = k0 + kk;
    v2f a;
    a.x = cosf(ts * basis_freq[k] + phase[k]);
    a.y = cosf(ts * basis_freq[k + 1] + phase[k + 1]);
#pragma unroll
    for (int t = 0; t < 8; ++t) {
      v2f b;
      const float* bp = W_time + (size_t)(DD + k) * DD + t * 16 + ncol;
      b.x = bp[0];
      b.y = bp[DD];
      acc[t] = wmma4(a, b, acc[t]);
    }
  }
  // Epilogue: bias + relu + store
  const int rowOut = rowBase + (lane >> 4) * 8;
#pragma unroll
  for (int t = 0; t < 8; ++t) {
    const int col = t * 16 + ncol;
    const float bias = b_time[col];
#pragma unroll
    for (int r = 0; r < 8; ++r) {
      float v = acc[t][r] + bias;
      h_self[(size_t)(rowOut + r) * DD + col] = fmaxf(v, 0.0f);
    }
  }
}

// ---------------------------------------------------------------------------
// Kernel 2: e = efeat @ W_edge + b_edge ; m = h_self[src] + e ;
//           neigh[dst] += m   (segment_sum via f32 atomics)
// One wave -> 16 edges x 128 cols, K = 64.
// W_edge (32 KB) staged in LDS via the Tensor Data Mover (one DMA per block,
// TENSORcnt-synchronized); falls back to vector loads when TDM builtin absent.
// ---------------------------------------------------------------------------
__global__ __launch_bounds__(256) void edge_kernel(
    const float* __restrict__ efeat, const float* __restrict__ W_edge,
    const float* __restrict__ b_edge, const float* __restrict__ h_self,
    const int* __restrict__ src, const int* __restrict__ dst,
    float* __restrict__ neigh) {
  __shared__ float sWe[EFD * DD];  // 64x128 f32 = 32 KB, LDS offset 0

#if USE_TDM
  // The TDM write to LDS is invisible to alias analysis (the descriptor only
  // carries the LDS address as an integer). Without a visible store, the
  // compiler folds all ds_loads of sWe to undef and dead-codes the WMMAs
  // (observed in codegen). Keep one volatile store of the SAME value the DMA
  // writes to that location: never elided, never forwarded, race-benign.
  if (threadIdx.x == 0) ((volatile float*)sWe)[0] = W_edge[0];
  if (threadIdx.x < 32) {  // one wave issues the DMA (EXEC ignored by TDM)
    const unsigned long long ga = (unsigned long long)(uintptr_t)W_edge;
    // D# group0: count=1 | lds_addr | global_addr[56:0] | type=2 ("image")
    u32x4 g0;
    g0.x = 0x1u;                 // count=1 valid user descriptor
    g0.y = 0u;                   // lds_addr: sWe is the only LDS alloc -> 0
    g0.z = (unsigned)(ga & 0xffffffffu);
    g0.w = (unsigned)((ga >> 32) & 0x1ffffffu) | (2u << 30);
    // D# group1: data_size=4B; 1-D tensor: dim0 = tile_dim0 = 8192 elements
    i32x8 g1;
    g1[0] = (2 << 16);                        // [17:16] data_size=2 (4 bytes)
    g1[1] = (int)((8192u & 0xffffu) << 16);   // tensor_dim0[15:0] -> [31:16]
    g1[2] = (int)((8192u >> 16) | (1u << 16));// tensor_dim0[31:16], dim1=1
    g1[3] = (int)(8192u << 16);               // tile_dim0=8192 -> [31:16]
    g1[4] = 1;                                // tile_dim1=1
    g1[5] = 8192;                             // tensor_dim0_stride (low 32)
    g1[6] = 0;
    g1[7] = 0;
    i32x4 z4 = {0, 0, 0, 0};
#if __clang_major__ >= 23
    i32x8 z8 = {0, 0, 0, 0, 0, 0, 0, 0};
    __builtin_amdgcn_tensor_load_to_lds(g0, g1, z4, z4, z8, 0);
#else
    __builtin_amdgcn_tensor_load_to_lds(g0, g1, z4, z4, 0);
#endif
    __builtin_amdgcn_s_wait_tensorcnt(0);
  }
  __syncthreads();
#else
  for (int i = threadIdx.x; i < EFD * DD; i += blockDim.x) sWe[i] = W_edge[i];
  __syncthreads();
#endif

  const int wave = (blockIdx.x * blockDim.x + threadIdx.x) >> 5;
  const int lane = threadIdx.x & 31;
  const int eBase = wave * 16;
  const int mrow = eBase + (lane & 15);
  const int kk = (lane >> 4) * 2;
  const int ncol = lane & 15;

  v8f acc[8] = {};
  for (int k0 = 0; k0 < EFD; k0 += 4) {
    v2f a;
    const float* ap = efeat + (size_t)mrow * EFD + k0 + kk;
    a.x = ap[0];
    a.y = ap[1];
#pragma unroll
    for (int t = 0; t < 8; ++t) {
      v2f b;
      const float* bp = sWe + (k0 + kk) * DD + t * 16 + ncol;
      b.x = bp[0];
      b.y = bp[DD];
      acc[t] = wmma4(a, b, acc[t]);
    }
  }

  // Epilogue: gather h_self[src], add bias, scatter-add into neigh[dst].
  const int rBase = eBase + (lane >> 4) * 8;
  int se[8], de[8];
#pragma unroll
  for (int r = 0; r < 8; ++r) {
    se[r] = src[rBase + r];
    de[r] = dst[rBase + r];
  }
#pragma unroll
  for (int t = 0; t < 8; ++t) {
    const int col = t * 16 + ncol;
    const float bias = b_edge[col];
#pragma unroll
    for (int r = 0; r < 8; ++r) {
      float m = acc[t][r] + bias + h_self[(size_t)se[r] * DD + col];
      unsafeAtomicAdd(&neigh[(size_t)de[r] * DD + col], m);  // global_atomic_add_f32
    }
  }
}

// ---------------------------------------------------------------------------
// Kernel 3: permuted segmented inclusive cumsum + scatter + degree divide.
// 64 blocks (one per group) x 1024 threads: col = tid&127, chunk = tid>>7.
// The per-chunk scans are serial chains of perm-indexed gathers, so issue
// global_prefetch_b8 for the row PF iterations ahead to hide gather latency.
// ---------------------------------------------------------------------------
#define PF 8
__global__ __launch_bounds__(1024) void scan_kernel(
    const float* __restrict__ neigh, const int* __restrict__ perm,
    const float* __restrict__ degree, float* __restrict__ h_neigh) {
  const int g = blockIdx.x;
  const int col = threadIdx.x & (DD - 1);
  const int chunk = threadIdx.x >> 7;  // 0..7
  const int rowsPerChunk = GSZ / 8;    // 256
  const int base = g * GSZ + chunk * rowsPerChunk;

  __shared__ float sums[8][DD];

  float s = 0.0f;
  for (int r = 0; r < rowsPerChunk; ++r) {
    const int rp = (r + PF < rowsPerChunk) ? r + PF : rowsPerChunk - 1;
    const int nodePf = perm[base + rp];
    __builtin_prefetch(&neigh[(size_t)nodePf * DD + col], 0, 1);
    const int node = perm[base + r];
    s += neigh[(size_t)node * DD + col];
  }
  sums[chunk][col] = s;
  __syncthreads();

  float off = 0.0f;
  for (int c = 0; c < chunk; ++c) off += sums[c][col];

  float acc = off;
  for (int r = 0; r < rowsPerChunk; ++r) {
    const int rp = (r + PF < rowsPerChunk) ? r + PF : rowsPerChunk - 1;
    const int nodePf = perm[base + rp];
    __builtin_prefetch(&neigh[(size_t)nodePf * DD + col], 0, 1);
    const int node = perm[base + r];
    acc += neigh[(size_t)node * DD + col];
    h_neigh[(size_t)node * DD + col] = acc / degree[node];
  }
}

// ---------------------------------------------------------------------------
// Kernel 4: out = h_self @ W_self + h_neigh @ W_neigh + (b_self + b_neigh)
// One wave -> 16 rows x 128 cols; two K=128 passes of f32 WMMA.
// ---------------------------------------------------------------------------
__global__ __launch_bounds__(256) void out_kernel(
    const float* __restrict__ h_self, const float* __restrict__ h_neigh,
    const float* __restrict__ W_self, const float* __restrict__ b_self,
    const float* __restrict__ W_neigh, const float* __restrict__ b_neigh,
    float* __restrict__ out) {
  const int wave = (blockIdx.x * blockDim.x + threadIdx.x) >> 5;
  const int lane = threadIdx.x & 31;
  const int rowBase = wave * 16;
  const int mrow = rowBase + (lane & 15);
  const int kk = (lane >> 4) * 2;
  const int ncol = lane & 15;

  v8f acc[8] = {};
  for (int k0 = 0; k0 < DD; k0 += 4) {
    v2f a;
    const float* ap = h_self + (size_t)mrow * DD + k0 + kk;
    a.x = ap[0];
    a.y = ap[1];
#pragma unroll
    for (int t = 0; t < 8; ++t) {
      v2f b;
      const float* bp = W_self + (size_t)(k0 + kk) * DD + t * 16 + ncol;
      b.x = bp[0];
      b.y = bp[DD];
      acc[t] = wmma4(a, b, acc[t]);
    }
  }
  for (int k0 = 0; k0 < DD; k0 += 4) {
    v2f a;
    const float* ap = h_neigh + (size_t)mrow * DD + k0 + kk;
    a.x = ap[0];
    a.y = ap[1];
#pragma unroll
    for (int t = 0; t < 8; ++t) {
      v2f b;
      const float* bp = W_neigh + (size_t)(k0 + kk) * DD + t * 16 + ncol;
      b.x = bp[0];
      b.y = bp[DD];
      acc[t] = wmma4(a, b, acc[t]);
    }
  }
  const int rowOut = rowBase + (lane >> 4) * 8;
#pragma unroll
  for (int t = 0; t < 8; ++t) {
    const int col = t * 16 + ncol;
    const float bias = b_self[col] + b_neigh[col];
#pragma unroll
    for (int r = 0; r < 8; ++r) {
      out[(size_t)(rowOut + r) * DD + col] = acc[t][r] + bias;
    }
  }
}

// ---------------------------------------------------------------------------
extern "C" void kernel_launch(void* const* d_in, const int* in_sizes, int n_in,
                              void* d_out, int out_size, void* d_ws,
                              size_t ws_size, hipStream_t stream) {
  const float* nfeat = (const float*)d_in[0];
  const float* timestamp = (const float*)d_in[1];
  const float* efeat = (const float*)d_in[2];
  const float* degree = (const float*)d_in[3];
  const int* src = (const int*)d_in[4];
  const int* dst = (const int*)d_in[5];
  const int* perm = (const int*)d_in[6];
  const float* basis_freq = (const float*)d_in[7];
  const float* phase = (const float*)d_in[8];
  const float* W_time = (const float*)d_in[9];
  const float* b_time = (const float*)d_in[10];
  const float* W_edge = (const float*)d_in[11];
  const float* b_edge = (const float*)d_in[12];
  const float* W_self = (const float*)d_in[13];
  const float* b_self = (const float*)d_in[14];
  const float* W_neigh = (const float*)d_in[15];
  const float* b_neigh = (const float*)d_in[16];

  // Workspace layout: h_self | neigh | h_neigh, each N*D f32 (64 MB) = 192 MB
  const size_t matBytes = (size_t)NN * DD * sizeof(float);
  float* h_self = (float*)d_ws;
  float* neigh = (float*)((char*)d_ws + matBytes);
  float* h_neigh = (float*)((char*)d_ws + 2 * matBytes);
  float* outp = (float*)d_out;

  // Zero the segment-sum accumulator (graph-capturable async memset).
  hipMemsetAsync(neigh, 0, matBytes, stream);

  // K1: h_self.  N/16 waves, 8 waves/block of 256 threads.
  {
    dim3 grid(NN / 16 / 8), block(256);
    hself_kernel<<<grid, block, 0, stream>>>(nfeat, timestamp, basis_freq,
                                             phase, W_time, b_time, h_self);
  }
  // K2: edge GEMM + gather + atomic segment-sum.  E/16 waves.
  {
    dim3 grid(EE / 16 / 8), block(256);
    edge_kernel<<<grid, block, 0, stream>>>(efeat, W_edge, b_edge, h_self, src,
                                            dst, neigh);
  }
  // K3: permuted segmented cumsum (64 groups).
  {
    dim3 grid(NGRP), block(1024);
    scan_kernel<<<grid, block, 0, stream>>>(neigh, perm, degree, h_neigh);
  }
  // K4: final fused dual GEMM.
  {
    dim3 grid(NN / 16 / 8), block(256);
    out_kernel<<<grid, block, 0, stream>>>(h_self, h_neigh, W_self, b_self,
                                           W_neigh, b_neigh, outp);
  }
}